// DistMultPredictor_86715389706550
// MI455X (gfx1250) — compile-verified
//
#include <hip/hip_runtime.h>

typedef float v2f __attribute__((ext_vector_type(2)));
typedef float v8f __attribute__((ext_vector_type(8)));

#define DIM 128
#define EXP_LAMBDA 0.7f
#define EPS 1e-8f

// ---------------------------------------------------------------------------
// Kernel 1: inverse (norm + eps) per disease row.  grid = N_DIS, block = 128.
// ---------------------------------------------------------------------------
__global__ __launch_bounds__(128) void norm_kernel(const float* __restrict__ h,
                                                   float* __restrict__ invn,
                                                   int n_dis) {
  const int j = blockIdx.x;
  if (j >= n_dis) return;
  const float x = h[(size_t)j * DIM + threadIdx.x];
  float p = x * x;
  p += __shfl_xor(p, 16);
  p += __shfl_xor(p, 8);
  p += __shfl_xor(p, 4);
  p += __shfl_xor(p, 2);
  p += __shfl_xor(p, 1);
  __shared__ float ws[4];
  if ((threadIdx.x & 31) == 0) ws[threadIdx.x >> 5] = p;
  __syncthreads();
  if (threadIdx.x == 0) {
    const float s = ws[0] + ws[1] + ws[2] + ws[3];
    invn[j] = 1.0f / (sqrtf(s) + EPS);
  }
}

// ---------------------------------------------------------------------------
// Kernel 2: drug-disease edges with prototype augmentation.
// One block (8 waves) handles 16 edges. The [16 x N_DIS] cosine-sim rows are
// produced tile-by-tile with V_WMMA_F32_16X16X4_F32 (fp32, K=4), with a
// streaming exact top-3 per edge.
// ---------------------------------------------------------------------------
__global__ __launch_bounds__(256) void dd_score_kernel(
    const float* __restrict__ h_drug, const float* __restrict__ h_dis,
    const float* __restrict__ W, const int* __restrict__ src_dd,
    const int* __restrict__ dst_dd, const int* __restrict__ deg_dis,
    const float* __restrict__ invn, float* __restrict__ out, int n_dis,
    int e_dd) {
  __shared__ float Asm[16 * 132];        // normalized dst rows, padded stride
  __shared__ float Csc[8][16][17];       // per-wave 16x16 C tile scratch
  __shared__ float TopV[8][16][3];
  __shared__ int   TopI[8][16][3];
  __shared__ int   sDst[16];
  __shared__ int   FinI[16][3];
  __shared__ float FinW[16][3];

  const int tid  = threadIdx.x;
  const int wave = tid >> 5;
  const int lane = tid & 31;
  const int e0   = blockIdx.x * 16;

  if (tid < 16) {
    const int e = e0 + tid;
    sDst[tid]   = dst_dd[(e < e_dd) ? e : (e_dd - 1)];
  }
  __syncthreads();

  // Stage A = normalized dst embeddings [16 x 128] into LDS.
  for (int i = tid; i < 16 * DIM; i += 256) {
    const int m = i >> 7;
    const int k = i & (DIM - 1);
    const int j = sDst[m];
    Asm[m * 132 + k] = h_dis[(size_t)j * DIM + k] * invn[j];
  }
  __syncthreads();

  // Per-wave A fragments: 32 K-steps of 16x4 fp32.
  const int mrow = lane & 15;
  const int koff = (lane < 16) ? 0 : 2;   // lanes 0-15: K=0,1  lanes 16-31: K=2,3
  v2f afrag[32];
#pragma unroll
  for (int kk = 0; kk < 32; ++kk) {
    const float2 t = *(const float2*)&Asm[mrow * 132 + kk * 4 + koff];
    afrag[kk].x = t.x;
    afrag[kk].y = t.y;
  }

  const float NEG = -__builtin_inff();
  float t0 = NEG, t1 = NEG, t2 = NEG;
  int   i0 = -1,  i1 = -1,  i2 = -1;
  const int selfJ = sDst[mrow];

  // Each wave owns disjoint 16-column stripes of the disease set.
  for (int col0 = wave * 16; col0 < n_dis; col0 += 128) {
    const int  j      = col0 + (lane & 15);
    const bool jvalid = (j < n_dis);
    const int  jj     = jvalid ? j : 0;
    const float inv   = jvalid ? invn[jj] : 0.0f;
    const float* bp   = h_dis + (size_t)jj * DIM + koff;

    v8f c = {0.f, 0.f, 0.f, 0.f, 0.f, 0.f, 0.f, 0.f};
#pragma unroll
    for (int kk = 0; kk < 32; ++kk) {
      const float2 bt = *(const float2*)(bp + kk * 4);
      v2f b;
      b.x = bt.x * inv;
      b.y = bt.y * inv;
      c = __builtin_amdgcn_wmma_f32_16x16x4_f32(false, afrag[kk], false, b,
                                                (short)0, c, false, false);
    }

    // C layout: VGPR g -> M = g (lanes 0-15) / g+8 (lanes 16-31), N = lane&15.
#pragma unroll
    for (int g = 0; g < 8; ++g) {
      const int m = (lane < 16) ? g : (g + 8);
      Csc[wave][m][lane & 15] = c[g];
    }
    // Wave-local scan: lane m (0..15) updates its edge's running top-3.
    if (lane < 16) {
      for (int n = 0; n < 16; ++n) {
        const int jc = col0 + n;
        if (jc >= n_dis || jc == selfJ) continue;
        const float v = Csc[wave][lane][n];
        if (v > t0) {
          t2 = t1; i2 = i1; t1 = t0; i1 = i0; t0 = v; i0 = jc;
        } else if (v > t1) {
          t2 = t1; i2 = i1; t1 = v; i1 = jc;
        } else if (v > t2) {
          t2 = v; i2 = jc;
        }
      }
    }
  }

  if (lane < 16) {
    TopV[wave][lane][0] = t0; TopV[wave][lane][1] = t1; TopV[wave][lane][2] = t2;
    TopI[wave][lane][0] = i0; TopI[wave][lane][1] = i1; TopI[wave][lane][2] = i2;
  }
  __syncthreads();

  // Merge 8 per-wave top-3 lists; compute softmax weights.
  if (wave == 0 && lane < 16) {
    float f0 = NEG, f1 = NEG, f2 = NEG;
    int   g0 = -1,  g1 = -1,  g2 = -1;
    for (int w2 = 0; w2 < 8; ++w2) {
      for (int r = 0; r < 3; ++r) {
        const float v  = TopV[w2][lane][r];
        const int   jc = TopI[w2][lane][r];
        if (jc < 0) continue;
        if (v > f0) {
          f2 = f1; g2 = g1; f1 = f0; g1 = g0; f0 = v; g0 = jc;
        } else if (v > f1) {
          f2 = f1; g2 = g1; f1 = v; g1 = jc;
        } else if (v > f2) {
          f2 = v; g2 = jc;
        }
      }
    }
    const float ea = expf(f0 - f0);
    const float eb = expf(f1 - f0);
    const float ec = expf(f2 - f0);
    const float si = 1.0f / (ea + eb + ec);
    FinW[lane][0] = ea * si; FinW[lane][1] = eb * si; FinW[lane][2] = ec * si;
    FinI[lane][0] = g0;      FinI[lane][1] = g1;      FinI[lane][2] = g2;
  }
  __syncthreads();

  // Final DistMult score: 16 threads per edge, 8 dims each.
  const int m    = tid >> 4;
  const int part = tid & 15;
  const int e    = e0 + m;
  float partial  = 0.0f;
  if (e < e_dd) {
    const int js   = src_dd[e];
    const int jd   = sDst[m];
    const float cf = expf(-EXP_LAMBDA * (float)deg_dis[jd]);
    const int   k0 = FinI[m][0], k1 = FinI[m][1], k2 = FinI[m][2];
    const float w0 = FinW[m][0], w1 = FinW[m][1], w2 = FinW[m][2];
    const float* ps = h_drug + (size_t)js * DIM + part * 8;
    const float* pd = h_dis  + (size_t)jd * DIM + part * 8;
    const float* p0 = h_dis  + (size_t)k0 * DIM + part * 8;
    const float* p1 = h_dis  + (size_t)k1 * DIM + part * 8;
    const float* p2 = h_dis  + (size_t)k2 * DIM + part * 8;
    const float* pw = W + part * 8;  // W row 0
#pragma unroll
    for (int r = 0; r < 2; ++r) {
      const float4 a  = *(const float4*)(ps + r * 4);
      const float4 dv = *(const float4*)(pd + r * 4);
      const float4 q0 = *(const float4*)(p0 + r * 4);
      const float4 q1 = *(const float4*)(p1 + r * 4);
      const float4 q2 = *(const float4*)(p2 + r * 4);
      const float4 wv = *(const float4*)(pw + r * 4);
      const float ax[4] = {a.x, a.y, a.z, a.w};
      const float dx[4] = {dv.x, dv.y, dv.z, dv.w};
      const float x0[4] = {q0.x, q0.y, q0.z, q0.w};
      const float x1[4] = {q1.x, q1.y, q1.z, q1.w};
      const float x2[4] = {q2.x, q2.y, q2.z, q2.w};
      const float wx[4] = {wv.x, wv.y, wv.z, wv.w};
#pragma unroll
      for (int u = 0; u < 4; ++u) {
        const float proto = w0 * x0[u] + w1 * x1[u] + w2 * x2[u];
        const float aug   = (1.0f - cf) * dx[u] + cf * proto;
        partial += ax[u] * wx[u] * aug;
      }
    }
  }
  partial += __shfl_xor(partial, 8);
  partial += __shfl_xor(partial, 4);
  partial += __shfl_xor(partial, 2);
  partial += __shfl_xor(partial, 1);
  if (part == 0 && e < e_dd) out[e] = partial;
}

// ---------------------------------------------------------------------------
// Kernel 3: disease-disease DistMult. Half-wave (16 lanes) per edge:
// coalesced float4 row reads (table is L2-resident), shuffle reduction.
// ---------------------------------------------------------------------------
__global__ __launch_bounds__(256) void nn_score_kernel(
    const float* __restrict__ h_dis, const float* __restrict__ W,
    const int* __restrict__ src_nn, const int* __restrict__ dst_nn,
    float* __restrict__ out, int e_nn) {
  const int tid  = threadIdx.x;
  const int part = tid & 15;
  const long e   = (long)blockIdx.x * 16 + (tid >> 4);
  float s = 0.0f;
  if (e < e_nn) {
    const int js = src_nn[e];
    const int jd = dst_nn[e];
    const float* ps = h_dis + (size_t)js * DIM + part * 8;
    const float* pd = h_dis + (size_t)jd * DIM + part * 8;
    const float* pw = W + DIM + part * 8;  // W row 1
#pragma unroll
    for (int r = 0; r < 2; ++r) {
      const float4 a = *(const float4*)(ps + r * 4);
      const float4 b = *(const float4*)(pd + r * 4);
      const float4 w = *(const float4*)(pw + r * 4);
      s += a.x * b.x * w.x + a.y * b.y * w.y + a.z * b.z * w.z + a.w * b.w * w.w;
    }
  }
  s += __shfl_xor(s, 8);
  s += __shfl_xor(s, 4);
  s += __shfl_xor(s, 2);
  s += __shfl_xor(s, 1);
  if (part == 0 && e < e_nn) out[e] = s;
}

// ---------------------------------------------------------------------------
extern "C" void kernel_launch(void* const* d_in, const int* in_sizes, int n_in,
                              void* d_out, int out_size, void* d_ws,
                              size_t ws_size, hipStream_t stream) {
  const float* h_drug = (const float*)d_in[0];
  const float* h_dis  = (const float*)d_in[1];
  const float* W      = (const float*)d_in[2];
  const int* src_dd   = (const int*)d_in[3];
  const int* dst_dd   = (const int*)d_in[4];
  const int* src_nn   = (const int*)d_in[5];
  const int* dst_nn   = (const int*)d_in[6];
  const int* deg_dis  = (const int*)d_in[7];

  const int n_dis = in_sizes[1] / DIM;
  const int e_dd  = in_sizes[3];
  const int e_nn  = in_sizes[5];

  float* out  = (float*)d_out;
  float* invn = (float*)d_ws;  // n_dis floats of scratch

  norm_kernel<<<n_dis, 128, 0, stream>>>(h_dis, invn, n_dis);
  dd_score_kernel<<<(e_dd + 15) / 16, 256, 0, stream>>>(
      h_drug, h_dis, W, src_dd, dst_dd, deg_dis, invn, out, n_dis, e_dd);
  nn_score_kernel<<<(e_nn + 15) / 16, 256, 0, stream>>>(
      h_dis, W, src_nn, dst_nn, out + e_dd, e_nn);
}